// CentroidHead_47313359732889
// MI455X (gfx1250) — compile-verified
//
#include <hip/hip_runtime.h>
#include <math.h>

#define NPTS   12288
#define DFEAT  128
#define TAUV   0.1f
#define R2V    1.21f
#define CHUNK  128

typedef __attribute__((ext_vector_type(2))) float v2f;
typedef __attribute__((ext_vector_type(8))) float v8f;

// ---------------------------------------------------------------------------
// Kernel 1: logits = feats @ W + b ; heat = sigmoid ; tag4 = {x,y,z, valid?b:-1}
// ---------------------------------------------------------------------------
__global__ __launch_bounds__(256) void ch_prep(
    const float* __restrict__ feats, const float* __restrict__ W,
    const float* __restrict__ bias, const int* __restrict__ coords,
    float* __restrict__ logits_out, float* __restrict__ heat_ws,
    float4* __restrict__ tag_ws) {
  __shared__ float Ws[DFEAT];
  const int t = threadIdx.x;
  if (t < DFEAT) Ws[t] = W[t];
  __syncthreads();
  const int i = blockIdx.x * 256 + t;
  const float* row = feats + (size_t)i * DFEAT;
  float acc = 0.0f;
#pragma unroll
  for (int k = 0; k < DFEAT; k += 4) {
    float4 f = *(const float4*)(row + k);
    acc = fmaf(f.x, Ws[k + 0], acc);
    acc = fmaf(f.y, Ws[k + 1], acc);
    acc = fmaf(f.z, Ws[k + 2], acc);
    acc = fmaf(f.w, Ws[k + 3], acc);
  }
  const float lg = acc + bias[0];
  logits_out[i] = lg;
  const float h = 1.0f / (1.0f + expf(-lg));
  heat_ws[i] = h;
  const int4 c = *(const int4*)(coords + (size_t)i * 4);  // {batch,x,y,z}
  float4 tg;
  tg.x = (float)c.y; tg.y = (float)c.z; tg.z = (float)c.w;
  tg.w = (h > TAUV) ? (float)c.x : -1.0f;  // fold valid[j] into batch tag
  tag_ws[i] = tg;
}

// ---------------------------------------------------------------------------
// Kernel 2: neigh_max over N x N (LDS tiled) -> is_peak ; peaks output
// ---------------------------------------------------------------------------
__global__ __launch_bounds__(256) void ch_peak(
    const float* __restrict__ heat_ws, const float4* __restrict__ tag_ws,
    const int* __restrict__ coords, float4* __restrict__ peaks_out,
    float* __restrict__ ispeak_ws) {
  __shared__ float4 stag[256];
  __shared__ float  sheat[256];
  const int t = threadIdx.x;
  const int i = blockIdx.x * 256 + t;
  const float4 ti = tag_ws[i];
  const float  hi = heat_ws[i];
  float nmax = -INFINITY;
  for (int base = 0; base < NPTS; base += 256) {
    __syncthreads();
    stag[t]  = tag_ws[base + t];
    sheat[t] = heat_ws[base + t];
    __syncthreads();
#pragma unroll 4
    for (int jj = 0; jj < 256; ++jj) {
      const float4 tj = stag[jj];
      const float dx = ti.x - tj.x, dy = ti.y - tj.y, dz = ti.z - tj.z;
      float d2 = dx * dx;
      d2 = fmaf(dy, dy, d2);
      d2 = fmaf(dz, dz, d2);
      const bool m = (ti.w == tj.w) && (d2 <= R2V);
      const float hj = sheat[jj];
      nmax = m ? fmaxf(nmax, hj) : nmax;
    }
  }
  const bool valid = (ti.w >= 0.0f);
  const bool peak = valid && (hi >= nmax);
  ispeak_ws[i] = peak ? 1.0f : 0.0f;
  const int4 c = *(const int4*)(coords + (size_t)i * 4);
  float4 pk;
  pk.x = peak ? (float)c.x : 0.0f;
  pk.y = peak ? (float)c.y : 0.0f;
  pk.z = peak ? (float)c.z : 0.0f;
  pk.w = peak ? (float)c.w : 0.0f;
  peaks_out[i] = pk;
}

// ---------------------------------------------------------------------------
// Kernel 3: feat_mean = (w @ feats) / cnt via V_WMMA_F32_16X16X4_F32
// One wave per 16-row i-tile, 8 f32 accumulators cover all 128 feature cols.
// ---------------------------------------------------------------------------
__global__ __launch_bounds__(256) void ch_cluster(
    const float* __restrict__ feats, const float4* __restrict__ tag_ws,
    const float* __restrict__ ispeak_ws, float* __restrict__ fm_out) {
  __shared__ float  sfeat[CHUNK * DFEAT];  // 64 KB feats chunk
  __shared__ float4 stag[CHUNK];           // 2 KB tags

  const int t    = threadIdx.x;
  const int wave = t >> 5;           // 8 waves / block (wave32)
  const int lane = t & 31;
  const int half = lane >> 4;        // 0: K={0,1}  1: K={2,3}
  const int mn   = lane & 15;        // A-row m  /  B-col n
  const int itile = blockIdx.x * 8 + wave;
  const int gi    = itile * 16 + mn;

  const float4 ti = tag_ws[gi];
  const float  pk = ispeak_ws[gi];
  // row tag: matches tag_j (= batch of a valid j) only if this row is a peak
  const float tagi = (pk > 0.5f) ? ti.w : -2.0f;

  v8f acc[8];
#pragma unroll
  for (int c = 0; c < 8; ++c) acc[c] = (v8f){0, 0, 0, 0, 0, 0, 0, 0};
  float cntp = 0.0f;

  for (int base = 0; base < NPTS; base += CHUNK) {
    __syncthreads();
    const float4* src = (const float4*)(feats + (size_t)base * DFEAT);
#pragma unroll
    for (int r = t; r < CHUNK * DFEAT / 4; r += 256) ((float4*)sfeat)[r] = src[r];
    if (t < CHUNK) stag[t] = tag_ws[base + t];
    __syncthreads();

    if (base + CHUNK < NPTS)  // speculative prefetch of the next chunk
      __builtin_prefetch(feats + (size_t)(base + CHUNK) * DFEAT + t * 64, 0, 0);

    for (int j = 0; j < CHUNK; j += 4) {
      const int k0 = j + 2 * half;
      const float4 t0 = stag[k0];
      const float4 t1 = stag[k0 + 1];
      float dx = ti.x - t0.x, dy = ti.y - t0.y, dz = ti.z - t0.z;
      float d2a = fmaf(dz, dz, fmaf(dy, dy, dx * dx));
      dx = ti.x - t1.x; dy = ti.y - t1.y; dz = ti.z - t1.z;
      float d2b = fmaf(dz, dz, fmaf(dy, dy, dx * dx));
      const float w0 = ((tagi == t0.w) && (d2a <= R2V)) ? 1.0f : 0.0f;
      const float w1 = ((tagi == t1.w) && (d2b <= R2V)) ? 1.0f : 0.0f;
      cntp += w0 + w1;
      const v2f a = {w0, w1};
      const float* fr0 = sfeat + k0 * DFEAT + mn;
#pragma unroll
      for (int c = 0; c < 8; ++c) {
        const v2f bm = {fr0[c * 16], fr0[DFEAT + c * 16]};
        acc[c] = __builtin_amdgcn_wmma_f32_16x16x4_f32(
            false, a, false, bm, (short)0, acc[c], false, false);
      }
    }
  }

  // cnt[row] = sum of both lane-halves; then every lane holds cnt[lane&15]
  const float cnt = cntp + __shfl_xor(cntp, 16, 32);

  float* outb = fm_out + (size_t)itile * 16 * DFEAT;
#pragma unroll
  for (int r = 0; r < 8; ++r) {
    const int row = r + 8 * half;                 // C/D layout: VGPR r, half
    const float cr = __shfl(cnt, row, 32);
    const float rcnt = 1.0f / fmaxf(cr, 1.0f);
#pragma unroll
    for (int c = 0; c < 8; ++c)
      outb[row * DFEAT + c * 16 + mn] = acc[c][r] * rcnt;
  }
}

// ---------------------------------------------------------------------------
extern "C" void kernel_launch(void* const* d_in, const int* in_sizes, int n_in,
                              void* d_out, int out_size, void* d_ws, size_t ws_size,
                              hipStream_t stream) {
  const float* feats  = (const float*)d_in[0];  // N*128
  const float* W      = (const float*)d_in[1];  // 128
  const float* bias   = (const float*)d_in[2];  // 1
  const int*   coords = (const int*)d_in[3];    // N*4

  float* out = (float*)d_out;
  float* logits_out = out;                  // N
  float4* peaks_out = (float4*)(out + NPTS);        // N*4
  float* fm_out     = out + NPTS + NPTS * 4;        // N*128

  float*  heat_ws   = (float*)d_ws;                 // N
  float4* tag_ws    = (float4*)(heat_ws + NPTS);    // N*4
  float*  ispeak_ws = (float*)(tag_ws + NPTS);      // N

  ch_prep<<<NPTS / 256, 256, 0, stream>>>(feats, W, bias, coords,
                                          logits_out, heat_ws, tag_ws);
  ch_peak<<<NPTS / 256, 256, 0, stream>>>(heat_ws, tag_ws, coords,
                                          peaks_out, ispeak_ws);
  ch_cluster<<<NPTS / (16 * 8), 256, 0, stream>>>(feats, tag_ws, ispeak_ws, fm_out);
}